// MGraph_26087631356275
// MI455X (gfx1250) — compile-verified
//
#include <hip/hip_runtime.h>
#include <math.h>

typedef __attribute__((ext_vector_type(2))) float v2f;
typedef __attribute__((ext_vector_type(4))) float v4f;
typedef __attribute__((ext_vector_type(8))) float v8f;

#define N_Q  4096
#define P_K  512
#define V_M  4096
#define H_D  256
#define TOPK 4
#define DIM  (N_Q + P_K + V_M)                  /* 8704 */
#define ADJ_ELEMS ((size_t)DIM * (size_t)DIM)   /* 75,759,616 */

__device__ __forceinline__ float wave_sum(float v) {
#pragma unroll
    for (int off = 16; off; off >>= 1) v += __shfl_xor(v, off, 32);
    return v;
}

__device__ __forceinline__ void nt_store4(float4* p, float4 v) {
    v4f w = {v.x, v.y, v.z, v.w};               // native clang vector for the builtin
    __builtin_nontemporal_store(w, (v4f*)p);
}

// ---------------------------------------------------------------------------
// 1) Row-normalize x (N rows) and vq_keys (P rows) into contiguous dst[(N+P)*H].
//    One wave32 per row; 2 float4 per lane (H=256).
// ---------------------------------------------------------------------------
__global__ void k_normalize(const float* __restrict__ x, const float* __restrict__ vq,
                            float* __restrict__ dst) {
    int row  = (int)((blockIdx.x * blockDim.x + threadIdx.x) >> 5);
    int lane = threadIdx.x & 31;
    if (row >= N_Q + P_K) return;
    const float* src = (row < N_Q) ? (x + (size_t)row * H_D)
                                   : (vq + (size_t)(row - N_Q) * H_D);
    float4 d0 = ((const float4*)src)[lane];
    float4 d1 = ((const float4*)src)[lane + 32];
    float ss = d0.x*d0.x + d0.y*d0.y + d0.z*d0.z + d0.w*d0.w
             + d1.x*d1.x + d1.y*d1.y + d1.z*d1.z + d1.w*d1.w;
    ss = wave_sum(ss);
    float s = 1.0f / fmaxf(sqrtf(ss), 1e-12f);
    float4* o = (float4*)(dst + (size_t)row * H_D);
    d0.x *= s; d0.y *= s; d0.z *= s; d0.w *= s;
    d1.x *= s; d1.y *= s; d1.z *= s; d1.w *= s;
    o[lane] = d0; o[lane + 32] = d1;
}

// ---------------------------------------------------------------------------
// 2) TF-IDF prep: scale[p] = 1/(sum_p(count)+1)  (wave per row)
// ---------------------------------------------------------------------------
__global__ void k_proto_scale(const float* __restrict__ pc, float* __restrict__ scale) {
    int p    = (int)((blockIdx.x * blockDim.x + threadIdx.x) >> 5);
    int lane = threadIdx.x & 31;
    if (p >= P_K) return;
    const float4* r = (const float4*)(pc + (size_t)p * V_M);
    float s = 0.f;
#pragma unroll 4
    for (int j = 0; j < (V_M / 4) / 32; ++j) {      // 32 iterations
        float4 c = r[lane + j * 32];
        s += c.x + c.y + c.z + c.w;
    }
    s = wave_sum(s);
    if (lane == 0) scale[p] = 1.0f / (s + 1.0f);
}

// ---------------------------------------------------------------------------
// 3) idf[v] = log((1+P)/(1+sum_motif[v])) + 1   (thread per column, coalesced)
// ---------------------------------------------------------------------------
__global__ void k_motif_idf(const float* __restrict__ pc, float* __restrict__ idf) {
    int v = blockIdx.x * blockDim.x + threadIdx.x;
    if (v >= V_M) return;
    float cnt = 0.f;
    for (int r = 0; r < P_K; ++r)
        cnt += (pc[(size_t)r * V_M + v] > 0.f) ? 1.f : 0.f;
    idf[v] = logf((1.f + (float)P_K) / (1.f + cnt)) + 1.f;
}

// ---------------------------------------------------------------------------
// 4) cos = Xn @ Kn^T via v_wmma_f32_16x16x4_f32.
//    Block = 8 waves = one 16-row strip; wave w covers protos [w*64, w*64+64).
// ---------------------------------------------------------------------------
__global__ void k_cos_wmma(const float* __restrict__ Xn, const float* __restrict__ Kn,
                           float* __restrict__ cosm) {
    const int tid   = threadIdx.x;
    const int wave  = tid >> 5;
    const int lane  = tid & 31;
    const int laneM = lane & 15;
    const int half  = lane >> 4;
    const int m0    = blockIdx.x * 16;

    const float* arow = Xn + (size_t)(m0 + laneM) * H_D + half * 2;

#pragma unroll
    for (int t = 0; t < 4; ++t) {
        const int p0 = wave * 64 + t * 16;
        const float* brow = Kn + (size_t)(p0 + laneM) * H_D + half * 2;
        v8f acc = {};
#pragma unroll 8
        for (int h = 0; h < H_D; h += 4) {
            v2f a = *(const v2f*)(arow + h);
            v2f b = *(const v2f*)(brow + h);
            acc = __builtin_amdgcn_wmma_f32_16x16x4_f32(
                false, a, false, b, (short)0, acc, false, false);
        }
        float* out = cosm + (size_t)m0 * P_K + p0 + laneM;
#pragma unroll
        for (int r = 0; r < 8; ++r)
            out[(size_t)(r + half * 8) * P_K] = acc[r];
    }
}

// ---------------------------------------------------------------------------
// 5) Top-4 per cos row (wave per row, 512 cols -> 16 per lane, 4 argmax passes;
//    ties resolve to the lower index, matching jax.lax.top_k).
// ---------------------------------------------------------------------------
__global__ void k_topk(const float* __restrict__ cosm, int* __restrict__ idx) {
    int row  = (int)((blockIdx.x * blockDim.x + threadIdx.x) >> 5);
    int lane = threadIdx.x & 31;
    if (row >= N_Q) return;
    const float* r = cosm + (size_t)row * P_K;
    float v[16];
#pragma unroll
    for (int j = 0; j < 16; ++j) v[j] = r[lane + j * 32];
#pragma unroll
    for (int k = 0; k < TOPK; ++k) {
        float bv = v[0]; int bc = lane;
#pragma unroll
        for (int j = 1; j < 16; ++j) {
            if (v[j] > bv) { bv = v[j]; bc = lane + j * 32; }
        }
#pragma unroll
        for (int off = 16; off; off >>= 1) {
            float ov = __shfl_xor(bv, off, 32);
            int   oc = __shfl_xor(bc, off, 32);
            if (ov > bv || (ov == bv && oc < bc)) { bv = ov; bc = oc; }
        }
        if (lane == 0) idx[row * TOPK + k] = bc;
        if ((bc & 31) == lane) v[bc >> 5] = -__builtin_inff();
    }
}

// ---------------------------------------------------------------------------
// 6) Fill the whole dense adjacency. Each thread writes 4 consecutive B128
//    non-temporal stores (64 B) in one row; zeros everywhere except the
//    TF-IDF block at rows [N, N+P) x cols [N+P, DIM). 303 MB stream > L2,
//    so NT policy avoids write-allocate churn and preserves pc/idf in cache.
//    DIM/16 = 544 chunks per row; block boundary col 4608 = chunk 288.
// ---------------------------------------------------------------------------
__global__ void k_fill(const float* __restrict__ pc, const float* __restrict__ scale,
                       const float* __restrict__ idf, float* __restrict__ adj) {
    size_t q16  = (size_t)blockIdx.x * blockDim.x + threadIdx.x;  // 16-float chunk id
    int row   = (int)(q16 / (DIM / 16));
    int col16 = (int)(q16 % (DIM / 16));
    float4* dst = (float4*)adj + q16 * 4;
    if (row >= N_Q && row < N_Q + P_K && col16 >= (N_Q + P_K) / 16) {
        int p  = row - N_Q;
        int v4 = col16 * 4 - (N_Q + P_K) / 4;       // first float4 index in pc/idf row
        float s = scale[p];
        const float4* cp = (const float4*)(pc + (size_t)p * V_M) + v4;
        const float4* fp = (const float4*)idf + v4;
#pragma unroll
        for (int j = 0; j < 4; ++j) {
            float4 c = cp[j];
            float4 f = fp[j];
            float4 val;
            val.x = c.x * s * f.x;
            val.y = c.y * s * f.y;
            val.z = c.z * s * f.z;
            val.w = c.w * s * f.w;
            nt_store4(dst + j, val);
        }
    } else {
        float4 z = make_float4(0.f, 0.f, 0.f, 0.f);
#pragma unroll
        for (int j = 0; j < 4; ++j) nt_store4(dst + j, z);
    }
}

// ---------------------------------------------------------------------------
// 7) Scatter the 16K top-k edges (after the fill).
// ---------------------------------------------------------------------------
__global__ void k_scatter(const int* __restrict__ idx, float* __restrict__ adj) {
    int t = blockIdx.x * blockDim.x + threadIdx.x;
    if (t >= N_Q * TOPK) return;
    int rowq = t >> 2;
    int col  = N_Q + idx[t];
    adj[(size_t)rowq * DIM + col] = 1.0f;
}

// ---------------------------------------------------------------------------
// 8) node_feat = concat(x, vq_keys)  (runs last; overwrites idx/scale/idf).
// ---------------------------------------------------------------------------
__global__ void k_feat(const float* __restrict__ x, const float* __restrict__ vq,
                       float* __restrict__ feat) {
    int q4 = blockIdx.x * blockDim.x + threadIdx.x;    // 294,912 total
    const int nx4 = N_Q * H_D / 4;
    float4 val = (q4 < nx4) ? ((const float4*)x)[q4]
                            : ((const float4*)vq)[q4 - nx4];
    nt_store4((float4*)feat + q4, val);
}

// ---------------------------------------------------------------------------
extern "C" void kernel_launch(void* const* d_in, const int* in_sizes, int n_in,
                              void* d_out, int out_size, void* d_ws, size_t ws_size,
                              hipStream_t stream) {
    (void)in_sizes; (void)n_in; (void)out_size; (void)d_ws; (void)ws_size;
    const float* x  = (const float*)d_in[0];   // [N, H]
    const float* vq = (const float*)d_in[1];   // [P, H]
    const float* pc = (const float*)d_in[2];   // [P, V]

    float* out  = (float*)d_out;
    float* adj  = out;                          // [DIM*DIM]
    float* feat = out + ADJ_ELEMS;              // [(N+P)*H]

    // scratch inside adj region (last read in k_topk, overwritten by k_fill):
    float* Xn   = adj;                                  // [(N+P)*H] (Xn then Kn)
    float* Kn   = adj + (size_t)N_Q * H_D;
    float* cosm = adj + (size_t)(N_Q + P_K) * H_D;      // [N*P]
    // scratch inside feat region (last read in k_scatter/k_fill, overwritten by k_feat):
    int*   idx   = (int*)feat;                          // [N*K]
    float* scale = feat + N_Q * TOPK;                   // [P]
    float* idf   = scale + P_K;                         // [V]

    k_normalize  <<<(N_Q + P_K) / 8,             256, 0, stream>>>(x, vq, Xn);
    k_proto_scale<<<P_K / 8,                     256, 0, stream>>>(pc, scale);
    k_motif_idf  <<<V_M / 256,                   256, 0, stream>>>(pc, idf);
    k_cos_wmma   <<<N_Q / 16,                    256, 0, stream>>>(Xn, Kn, cosm);
    k_topk       <<<N_Q / 8,                     256, 0, stream>>>(cosm, idx);
    k_fill       <<<(unsigned)(ADJ_ELEMS / 16 / 256), 256, 0, stream>>>(pc, scale, idf, adj);
    k_scatter    <<<(N_Q * TOPK) / 256,          256, 0, stream>>>(idx, adj);
    k_feat       <<<(N_Q + P_K) * H_D / 4 / 256, 256, 0, stream>>>(x, vq, feat);
}